// RotaryEmbeddingWithCache_4964982194249
// MI455X (gfx1250) — compile-verified
//
#include <hip/hip_runtime.h>

// Rotary-embedding cache gather:
//   out[row][0..127] = freqs_table[min(pos[row], 8191)][0..127]
// rows = 32*8192 = 262144, each row = 64*(cos,sin) = 128 f32 = 512 bytes.
//
// MI455X (wave32) strategy:
//   - One wave moves ROWS_PER_WAVE rows; lane L moves float4 L of each row
//     (32 lanes x 16B = 512B = one row per global_load_b128 / store_b128).
//   - Positions are wave-uniform -> readfirstlane'd index + const/restrict
//     lowers them to clause-able s_load_b32 (SMEM, single s_wait_kmcnt).
//   - Branch-free fast path: all 4 scalar position loads issue together,
//     then 4 independent global_load_b128, then 4 NT global_store_b128.
//   - Output is write-once: non-temporal stores keep the 128 MiB stream from
//     evicting the 4 MiB table out of WGP$/L2 (table loads stay RT).
//   - global_prefetch_b8 warms the positions stream (gfx1250 prefetch path).

typedef float v4f __attribute__((ext_vector_type(4)));

#define MAX_POS        8191
#define ROW_F4         32        // 128 floats per row = 32 float4
#define ROWS_PER_WAVE  4
#define BLOCK_THREADS  256       // 8 waves per block (wave32)

__global__ __launch_bounds__(BLOCK_THREADS)
void rope_cache_gather_kernel(const int* __restrict__ pos,      // [nrows]
                              const v4f* __restrict__ table,    // [8192][32]
                              v4f* __restrict__ out,            // [nrows][32]
                              int nrows)
{
    const int lane = threadIdx.x & 31;

    // Wave index, forced wave-uniform so downstream address math is scalar.
    int wave = (int)(blockIdx.x * (blockDim.x >> 5)) + (int)(threadIdx.x >> 5);
    wave = __builtin_amdgcn_readfirstlane(wave);
    const int row0 = wave * ROWS_PER_WAVE;

    // Prefetch the positions stream ahead of this block (global_prefetch_b8).
    if (lane == 0) {
        __builtin_prefetch(pos + row0 + ROWS_PER_WAVE * (BLOCK_THREADS / 32), 0, 3);
    }

    if (row0 + ROWS_PER_WAVE <= nrows) {
        // ---- Fast path: branch-free, fully pipelined ----
        // 1) All position reads issue back-to-back (scalar clause, one kmcnt wait).
        int p[ROWS_PER_WAVE];
#pragma unroll
        for (int i = 0; i < ROWS_PER_WAVE; ++i) {
            p[i] = pos[row0 + i];                       // wave-uniform -> s_load_b32
        }
#pragma unroll
        for (int i = 0; i < ROWS_PER_WAVE; ++i) {
            p[i] = (p[i] > MAX_POS) ? MAX_POS : p[i];   // jnp.clip(a_max=8191)
        }

        // 2) Four independent 512B row gathers in flight (global_load_b128).
        v4f v[ROWS_PER_WAVE];
#pragma unroll
        for (int i = 0; i < ROWS_PER_WAVE; ++i) {
            v[i] = table[(size_t)p[i] * ROW_F4 + lane];
        }

        // 3) Streaming stores (global_store_b128 th:TH_STORE_NT).
#pragma unroll
        for (int i = 0; i < ROWS_PER_WAVE; ++i) {
            __builtin_nontemporal_store(v[i], &out[(size_t)(row0 + i) * ROW_F4 + lane]);
        }
    } else {
        // ---- Tail path (unused for 262144 rows; kept for generality) ----
        for (int row = row0; row < nrows; ++row) {
            int pp = pos[row];
            pp = (pp > MAX_POS) ? MAX_POS : pp;
            const v4f v = table[(size_t)pp * ROW_F4 + lane];
            __builtin_nontemporal_store(v, &out[(size_t)row * ROW_F4 + lane]);
        }
    }
}

extern "C" void kernel_launch(void* const* d_in, const int* in_sizes, int n_in,
                              void* d_out, int out_size, void* d_ws, size_t ws_size,
                              hipStream_t stream) {
    (void)n_in; (void)d_ws; (void)ws_size; (void)out_size;

    const int* pos   = (const int*)d_in[0];   // positions, int32, 32*8192 elements
    const v4f* table = (const v4f*)d_in[1];   // freqs_table, f32, 8192*64*2 elements
    v4f*       out   = (v4f*)d_out;           // f32, 32*8192*64*2 elements

    const int nrows          = in_sizes[0];                     // 262144
    const int waves          = (nrows + ROWS_PER_WAVE - 1) / ROWS_PER_WAVE;
    const int waves_per_blk  = BLOCK_THREADS / 32;
    const int blocks         = (waves + waves_per_blk - 1) / waves_per_blk;

    rope_cache_gather_kernel<<<blocks, BLOCK_THREADS, 0, stream>>>(pos, table, out, nrows);
}